// CVLoss_23940147708353
// MI455X (gfx1250) — compile-verified
//
#include <hip/hip_runtime.h>
#include <math.h>

// Problem constants (match reference)
#define NB 16
#define NT 2000
#define NN 512
#define NL (NB * NT)        // 32000 flat time steps per neuron
#define NCHUNK 256
#define CLEN 125            // 256 * 125 = 32000
#define K1_THREADS 128      // 4 neurons per thread (float4)
#define STAGE 5             // time steps per async stage
#define NSTAGES (CLEN / STAGE)

#ifndef USE_ASYNC_LDS
#define USE_ASYNC_LDS 1
#endif

#if USE_ASYNC_LDS && defined(__AMDGCN__) && \
    __has_builtin(__builtin_amdgcn_global_load_async_to_lds_b128) && \
    __has_builtin(__builtin_amdgcn_s_wait_asynccnt)
#define ASYNC_PATH 1
#else
#define ASYNC_PATH 0
#endif

#if ASYNC_PATH
// Builtin signature (from clang diagnostics):
//   void __builtin_amdgcn_global_load_async_to_lds_b128(
//       v4i addrspace(1)* gsrc, v4i addrspace(3)* ldst, imm offset, imm cpol)
typedef int v4i __attribute__((ext_vector_type(4)));
typedef __attribute__((address_space(1))) v4i GVec;
typedef __attribute__((address_space(3))) v4i LVec;
#endif

// Per-(chunk,neuron) partial: x=count, y=first spike idx, z=last spike idx, w=sum d^2
__device__ __forceinline__ void upd_neuron(float xv, int i,
                                           int& cnt, int& first, int& last, int& ssq) {
    if (xv > 0.0f) {
        if (cnt > 0) { int d = i - last; ssq += d * d; }
        else         { first = i; }
        last = i;
        cnt += 1;
    }
}

__global__ __launch_bounds__(K1_THREADS)
void cv_k1_chunk_scan(const float* __restrict__ spikes, int4* __restrict__ partials) {
    const int c   = blockIdx.x;      // chunk id 0..255
    const int tid = threadIdx.x;     // neuron group: neurons 4*tid .. 4*tid+3
    const int base_i = c * CLEN;

    int cnt[4]   = {0, 0, 0, 0};
    int first[4] = {0, 0, 0, 0};
    int last[4]  = {0, 0, 0, 0};
    int ssq[4]   = {0, 0, 0, 0};

#if ASYNC_PATH
    // Double-buffered async global->LDS stream: each lane loads its own float4
    // (4 neurons x 1 step) into its private LDS slot; consumed by the same lane,
    // so only s_wait_asynccnt ordering is required (no barriers).
    __shared__ float4 abuf[2][STAGE * K1_THREADS];     // 20 KB
    const char* gbase = (const char*)spikes;

    // issue stage 0
    {
        const int pb = 0;
        for (int u = 0; u < STAGE; ++u) {
            size_t goff = ((size_t)(base_i + u) * NN + (size_t)tid * 4) * sizeof(float);
            __builtin_amdgcn_global_load_async_to_lds_b128(
                (GVec*)(gbase + goff),
                (LVec*)&abuf[pb][u * K1_THREADS + tid], 0, 0);
        }
    }
    for (int s = 0; s < NSTAGES; ++s) {
        if (s + 1 < NSTAGES) {
            const int pb = (s + 1) & 1;
            for (int u = 0; u < STAGE; ++u) {
                size_t goff = ((size_t)(base_i + (s + 1) * STAGE + u) * NN
                               + (size_t)tid * 4) * sizeof(float);
                __builtin_amdgcn_global_load_async_to_lds_b128(
                    (GVec*)(gbase + goff),
                    (LVec*)&abuf[pb][u * K1_THREADS + tid], 0, 0);
            }
            __builtin_amdgcn_s_wait_asynccnt(STAGE);   // stage s complete, s+1 in flight
        } else {
            __builtin_amdgcn_s_wait_asynccnt(0);       // final stage complete
        }
        const int rb = s & 1;
        for (int u = 0; u < STAGE; ++u) {
            float4 v = abuf[rb][u * K1_THREADS + tid];
            int i = base_i + s * STAGE + u;
            upd_neuron(v.x, i, cnt[0], first[0], last[0], ssq[0]);
            upd_neuron(v.y, i, cnt[1], first[1], last[1], ssq[1]);
            upd_neuron(v.z, i, cnt[2], first[2], last[2], ssq[2]);
            upd_neuron(v.w, i, cnt[3], first[3], last[3], ssq[3]);
        }
    }
#else
    // Fallback: direct coalesced float4 loads + gfx1250 global_prefetch ahead.
    const float4* p = (const float4*)spikes;           // NN/4 float4 per step
    const size_t col = (size_t)tid;                    // float4 column
    for (int r = 0; r < CLEN; ++r) {
        int i = base_i + r;
        if (r + 16 < CLEN) {
            __builtin_prefetch((const void*)&p[(size_t)(i + 16) * (NN / 4) + col], 0, 1);
        }
        float4 v = p[(size_t)i * (NN / 4) + col];
        upd_neuron(v.x, i, cnt[0], first[0], last[0], ssq[0]);
        upd_neuron(v.y, i, cnt[1], first[1], last[1], ssq[1]);
        upd_neuron(v.z, i, cnt[2], first[2], last[2], ssq[2]);
        upd_neuron(v.w, i, cnt[3], first[3], last[3], ssq[3]);
    }
#endif

    // partials layout: [chunk][neuron] -> coalesced stores here
    int4* out = partials + (size_t)c * NN + (size_t)tid * 4;
    out[0] = make_int4(cnt[0], first[0], last[0], ssq[0]);
    out[1] = make_int4(cnt[1], first[1], last[1], ssq[1]);
    out[2] = make_int4(cnt[2], first[2], last[2], ssq[2]);
    out[3] = make_int4(cnt[3], first[3], last[3], ssq[3]);
}

// Associative, ordered combine of adjacent time segments.
__device__ __forceinline__ int4 seg_comb(int4 a, int4 b) {
    if (a.x == 0) return b;
    if (b.x == 0) return a;
    int d = b.y - a.z;                       // gap between segments
    return make_int4(a.x + b.x, a.y, b.z, a.w + b.w + d * d);
}

__global__ __launch_bounds__(128)
void cv_k2_combine(const int4* __restrict__ partials, const float* __restrict__ target,
                   float* __restrict__ sq_out, int* __restrict__ valid_out) {
    const int n = blockIdx.x;                // neuron
    const int j = threadIdx.x;               // 0..127, owns chunks 2j, 2j+1
    __shared__ int4 seg[128];

    int4 a = partials[(size_t)(2 * j) * NN + n];
    int4 b = partials[(size_t)(2 * j + 1) * NN + n];
    seg[j] = seg_comb(a, b);
    __syncthreads();

    for (int m = 64; m >= 1; m >>= 1) {      // ordered pairwise tree
        int4 v;
        if (j < m) v = seg_comb(seg[2 * j], seg[2 * j + 1]);
        __syncthreads();
        if (j < m) seg[j] = v;
        __syncthreads();
    }

    if (j == 0) {
        int4 t = seg[0];
        int k = t.x;
        bool valid = (k >= 3);               // k>=3 implies last>first, mean>0
        float sq = 0.0f;
        if (valid) {
            int cnt = k - 1;                                     // #ISIs
            int sumd = t.z - t.y;                                // telescoped sum of ISIs
            float mean = (float)sumd / (float)cnt;               // matches f32 reference
            double md = (double)mean;
            double var_d = ((double)t.w - 2.0 * md * (double)sumd
                            + (double)cnt * md * md) / (double)((cnt - 1 > 1) ? cnt - 1 : 1);
            float var = (float)var_d;
            float stdv = (var > 0.0f) ? sqrtf(var) : 0.0f;
            float cv = stdv / fmaxf(mean, 1e-12f);
            float e = cv - target[n];
            sq = e * e;
        }
        sq_out[n] = sq;
        valid_out[n] = valid ? 1 : 0;
    }
}

__global__ __launch_bounds__(NN)
void cv_k3_loss(const float* __restrict__ sq, const int* __restrict__ valid,
                float* __restrict__ out) {
    __shared__ float ss[NN];
    __shared__ int sc[NN];
    const int t = threadIdx.x;
    ss[t] = sq[t];
    sc[t] = valid[t];
    __syncthreads();
    for (int m = NN / 2; m >= 1; m >>= 1) {
        if (t < m) { ss[t] += ss[t + m]; sc[t] += sc[t + m]; }
        __syncthreads();
    }
    if (t == 0) {
        int nv = sc[0] > 1 ? sc[0] : 1;
        out[0] = ss[0] / (float)nv;
    }
}

extern "C" void kernel_launch(void* const* d_in, const int* in_sizes, int n_in,
                              void* d_out, int out_size, void* d_ws, size_t ws_size,
                              hipStream_t stream) {
    (void)in_sizes; (void)n_in; (void)out_size; (void)ws_size;
    const float* spikes = (const float*)d_in[0];   // (16, 2000, 512) f32
    const float* target = (const float*)d_in[1];   // (512,) f32
    float* out = (float*)d_out;                    // scalar f32 loss

    // workspace layout
    int4*  partials = (int4*)d_ws;                                  // 256*512*16B = 2 MB
    char*  ws2      = (char*)d_ws + (size_t)NCHUNK * NN * sizeof(int4);
    float* sq_out   = (float*)ws2;                                  // 512 f32
    int*   vld_out  = (int*)(ws2 + NN * sizeof(float));             // 512 i32

    cv_k1_chunk_scan<<<NCHUNK, K1_THREADS, 0, stream>>>(spikes, partials);
    cv_k2_combine<<<NN, 128, 0, stream>>>(partials, target, sq_out, vld_out);
    cv_k3_loss<<<1, NN, 0, stream>>>(sq_out, vld_out, out);
}